// GAT_57509612093518
// MI455X (gfx1250) — compile-verified
//
#include <hip/hip_runtime.h>

typedef __attribute__((ext_vector_type(2))) float v2f;
typedef __attribute__((ext_vector_type(8))) float v8f;

#define LDS_PAD 4

// ---------------------------------------------------------------------------
// GEMM: H[N,Fout] = X[N,Fin] @ W[Fin,Fout] in fp32 via V_WMMA_F32_16X16X4_F32.
// Block = 256 threads (8 waves). blockIdx.y selects a 16-row strip of X
// (staged once in LDS, shared by all 8 waves); each wave owns one 16-col tile.
// ISA f32 layouts (CDNA5 ISA §7.12.2):
//   A 16x4 : lane L, vgpr j -> A[L&15][2*(L>>4)+j]
//   B 4x16 : lane L, vgpr j -> B[2*(L>>4)+j][L&15]
//   C 16x16: lane L, vgpr j -> C[j+8*(L>>4)][L&15]
// ---------------------------------------------------------------------------
template <int FIN>
__global__ __launch_bounds__(256) void gat_gemm_wmma_f32(
    const float* __restrict__ X, const float* __restrict__ W,
    float* __restrict__ H, int Fout) {
  __shared__ float ldsA[16 * (FIN + LDS_PAD)];
  const int m0   = blockIdx.y * 16;
  const int wave = threadIdx.x >> 5;
  const int lane = threadIdx.x & 31;
  const int n0   = (blockIdx.x * 8 + wave) * 16;

  for (int i = threadIdx.x; i < 16 * FIN; i += 256) {
    const int r = i / FIN, c = i % FIN;
    ldsA[r * (FIN + LDS_PAD) + c] = X[(size_t)(m0 + r) * FIN + c];
  }
  __syncthreads();

  const int mr = lane & 15;
  const int kh = lane >> 4;
  const float* ldsRow = &ldsA[mr * (FIN + LDS_PAD)];

  v8f acc = {};
  for (int k = 0; k < FIN; k += 4) {
    v2f a, b;
    a[0] = ldsRow[k + 2 * kh + 0];
    a[1] = ldsRow[k + 2 * kh + 1];
    b[0] = W[(size_t)(k + 2 * kh + 0) * Fout + n0 + mr];
    b[1] = W[(size_t)(k + 2 * kh + 1) * Fout + n0 + mr];
    acc = __builtin_amdgcn_wmma_f32_16x16x4_f32(
        false, a, false, b, (short)0, acc, false, false);
  }

#pragma unroll
  for (int j = 0; j < 8; ++j) {
    H[(size_t)(m0 + j + 8 * kh) * Fout + n0 + mr] = acc[j];
  }
}

// ---------------------------------------------------------------------------
// Per-node attention logits: e_src[n,h] = <H[n,h,:], a_src[h,:]>, same for dst.
// ---------------------------------------------------------------------------
__global__ void gat_att_dots(const float* __restrict__ H,
                             const float* __restrict__ a_src,
                             const float* __restrict__ a_dst,
                             float* __restrict__ es, float* __restrict__ ed,
                             int N, int heads, int ch) {
  const int idx = blockIdx.x * blockDim.x + threadIdx.x;
  if (idx >= N * heads) return;
  const int n = idx / heads, h = idx % heads;
  const float* row = H + (size_t)n * heads * ch + (size_t)h * ch;
  const float* as  = a_src + h * ch;
  const float* ad  = a_dst + h * ch;
  float s0 = 0.f, s1 = 0.f;
  for (int c = 0; c < ch; ++c) {
    const float v = row[c];
    s0 += v * as[c];
    s1 += v * ad[c];
  }
  es[idx] = s0;
  ed[idx] = s1;
}

// ---------------------------------------------------------------------------
// CSR build over destinations (shared by both layers; self-loops appended as
// edge ids >= E with src = dst = id - E).
// ---------------------------------------------------------------------------
__global__ void gat_zero2_int(int* __restrict__ a, int* __restrict__ b, int n) {
  const int i = blockIdx.x * blockDim.x + threadIdx.x;
  if (i < n) { a[i] = 0; b[i] = 0; }
}

__global__ void gat_csr_count(const int* __restrict__ dst, int E, int N,
                              int* __restrict__ counts) {
  const int idx = blockIdx.x * blockDim.x + threadIdx.x;
  if (idx >= E + N) return;
  const int d = (idx < E) ? dst[idx] : (idx - E);
  atomicAdd(&counts[d], 1);
}

// Single-block chunked exclusive scan (N = 50000 -> ~49 chunks of 1024).
__global__ __launch_bounds__(1024) void gat_scan(const int* __restrict__ counts,
                                                 int* __restrict__ offsets,
                                                 int N) {
  __shared__ int lds[1024];
  __shared__ int carry;
  if (threadIdx.x == 0) carry = 0;
  __syncthreads();
  for (int base = 0; base < N; base += 1024) {
    const int i = base + (int)threadIdx.x;
    const int v = (i < N) ? counts[i] : 0;
    lds[threadIdx.x] = v;
    __syncthreads();
    for (int off = 1; off < 1024; off <<= 1) {   // Hillis-Steele inclusive
      const int t = (threadIdx.x >= (unsigned)off) ? lds[threadIdx.x - off] : 0;
      __syncthreads();
      lds[threadIdx.x] += t;
      __syncthreads();
    }
    if (i < N) offsets[i] = carry + lds[threadIdx.x] - v;  // exclusive
    __syncthreads();
    if (threadIdx.x == 0) carry += lds[1023];
    __syncthreads();
  }
  if (threadIdx.x == 0) offsets[N] = carry;
}

__global__ void gat_csr_fill(const int* __restrict__ dst, int E, int N,
                             const int* __restrict__ offsets,
                             int* __restrict__ cursor, int* __restrict__ eids) {
  const int idx = blockIdx.x * blockDim.x + threadIdx.x;
  if (idx >= E + N) return;
  const int d = (idx < E) ? dst[idx] : (idx - E);
  const int pos = offsets[d] + atomicAdd(&cursor[d], 1);
  eids[pos] = idx;
}

// ---------------------------------------------------------------------------
// Fused per-node segment softmax + weighted aggregation + bias + ELU.
// One 256-thread block per destination node; F = HEADS*CH = 256.
// Phase 1: per-head max over incoming edges (LDS tree reduce; the stride stays
//          a multiple of HEADS so head classes never mix).
// Phase 2: per-head sum of exp(v - max).
// Phase 3: chunked aggregation: 256/HEADS edges' alphas computed in parallel
//          into LDS, then all 256 threads (thread = feature) accumulate
//          coalesced 1KB rows of H[src,:]. No global atomics anywhere.
// ---------------------------------------------------------------------------
template <int HEADS, int CH>
__global__ __launch_bounds__(256) void gat_node_softmax_agg(
    const int* __restrict__ offsets, const int* __restrict__ eids,
    const int* __restrict__ src, int E,
    const float* __restrict__ es, const float* __restrict__ ed,
    const float* __restrict__ H, const float* __restrict__ bias,
    float* __restrict__ out) {
  constexpr int F   = HEADS * CH;   // 256
  constexpr int CPE = 256 / HEADS;  // edges per chunk in phase 3
  const int n = blockIdx.x;
  const int t = threadIdx.x;
  const int beg = offsets[n];
  const int deg = offsets[n + 1] - beg;

  __shared__ float red[256];
  __shared__ float hmax[HEADS];
  __shared__ float hsum[HEADS];
  __shared__ float alph[256];
  __shared__ int   sidx[CPE];

  const int hc = t % HEADS;  // head class is invariant under +=256 striding

  // ---- phase 1: per-head max ----
  float lmax = -__builtin_inff();
  for (int q = t; q < deg * HEADS; q += 256) {
    const int i = q / HEADS;
    const int e = eids[beg + i];
    const int s = (e < E) ? src[e] : (e - E);
    float v = es[s * HEADS + hc] + ed[n * HEADS + hc];
    v = (v > 0.f) ? v : 0.2f * v;
    lmax = fmaxf(lmax, v);
  }
  red[t] = lmax;
  __syncthreads();
  for (int off = 128; off >= HEADS; off >>= 1) {
    if (t < off) red[t] = fmaxf(red[t], red[t + off]);
    __syncthreads();
  }
  if (t < HEADS) hmax[t] = red[t];
  __syncthreads();

  // ---- phase 2: per-head sum of exp ----
  float lsum = 0.f;
  for (int q = t; q < deg * HEADS; q += 256) {
    const int i = q / HEADS;
    const int e = eids[beg + i];
    const int s = (e < E) ? src[e] : (e - E);
    float v = es[s * HEADS + hc] + ed[n * HEADS + hc];
    v = (v > 0.f) ? v : 0.2f * v;
    lsum += expf(v - hmax[hc]);
  }
  red[t] = lsum;
  __syncthreads();
  for (int off = 128; off >= HEADS; off >>= 1) {
    if (t < off) red[t] += red[t + off];
    __syncthreads();
  }
  if (t < HEADS) hsum[t] = red[t];
  __syncthreads();

  // ---- phase 3: weighted aggregation (thread = feature) ----
  float acc = 0.f;
  const int hf = t / CH;  // head of this feature channel
  for (int c0 = 0; c0 < deg; c0 += CPE) {
    const int i = c0 + t / HEADS;
    if (i < deg) {
      const int e = eids[beg + i];
      const int s = (e < E) ? src[e] : (e - E);
      if (hc == 0) sidx[t / HEADS] = s;
      float v = es[s * HEADS + hc] + ed[n * HEADS + hc];
      v = (v > 0.f) ? v : 0.2f * v;
      alph[t] = expf(v - hmax[hc]) / hsum[hc];
    }
    __syncthreads();
    const int lim = min(CPE, deg - c0);
    for (int j = 0; j < lim; ++j) {
      acc += H[(size_t)sidx[j] * F + t] * alph[j * HEADS + hf];
    }
    __syncthreads();
  }

  const float v = acc + bias[t];
  out[(size_t)n * F + t] = (v > 0.f) ? v : (expf(v) - 1.f);
}

// ---------------------------------------------------------------------------
extern "C" void kernel_launch(void* const* d_in, const int* in_sizes, int n_in,
                              void* d_out, int out_size, void* d_ws,
                              size_t ws_size, hipStream_t stream) {
  const float* x   = (const float*)d_in[0];
  const int*   ei  = (const int*)d_in[1];
  const float* W1  = (const float*)d_in[2];
  const float* as1 = (const float*)d_in[3];
  const float* ad1 = (const float*)d_in[4];
  const float* b1  = (const float*)d_in[5];
  const float* W2  = (const float*)d_in[6];
  const float* as2 = (const float*)d_in[7];
  const float* ad2 = (const float*)d_in[8];
  const float* b2  = (const float*)d_in[9];

  const int N = in_sizes[0] / 128;  // 50000
  const int E = in_sizes[1] / 2;    // 800000
  const int* src = ei;
  const int* dst = ei + E;
  const int Etot = E + N;

  // workspace carve-up (all 4-byte elements)
  float* ws = (float*)d_ws;
  size_t o = 0;
  float* h    = ws + o; o += (size_t)N * 256;
  float* z1   = ws + o; o += (size_t)N * 256;
  float* es   = ws + o; o += (size_t)N * 8;
  float* edb  = ws + o; o += (size_t)N * 8;
  int* counts  = (int*)(ws + o); o += (size_t)N;
  int* cursor  = (int*)(ws + o); o += (size_t)N;
  int* offsets = (int*)(ws + o); o += (size_t)N + 1;
  int* eids    = (int*)(ws + o); o += (size_t)Etot;
  (void)ws_size; (void)n_in; (void)out_size;

  // ---- CSR over destinations (used by both layers) ----
  gat_zero2_int<<<(N + 255) / 256, 256, 0, stream>>>(counts, cursor, N);
  gat_csr_count<<<(Etot + 255) / 256, 256, 0, stream>>>(dst, E, N, counts);
  gat_scan<<<1, 1024, 0, stream>>>(counts, offsets, N);
  gat_csr_fill<<<(Etot + 255) / 256, 256, 0, stream>>>(dst, E, N, offsets,
                                                       cursor, eids);

  // ---- layer 1: heads=8, ch=32, Fin=128 ----
  {
    dim3 gg(256 / 16 / 8, N / 16);
    gat_gemm_wmma_f32<128><<<gg, 256, 0, stream>>>(x, W1, h, 256);
    gat_att_dots<<<(N * 8 + 255) / 256, 256, 0, stream>>>(h, as1, ad1, es, edb,
                                                          N, 8, 32);
    gat_node_softmax_agg<8, 32><<<N, 256, 0, stream>>>(offsets, eids, src, E,
                                                       es, edb, h, b1, z1);
  }

  // ---- layer 2: heads=1, ch=256, Fin=256 ----
  {
    dim3 gg(256 / 16 / 8, N / 16);
    gat_gemm_wmma_f32<256><<<gg, 256, 0, stream>>>(z1, W2, h, 256);
    gat_att_dots<<<(N + 255) / 256, 256, 0, stream>>>(h, as2, ad2, es, edb, N,
                                                      1, 256);
    gat_node_softmax_agg<1, 256><<<N, 256, 0, stream>>>(offsets, eids, src, E,
                                                        es, edb, h, b2,
                                                        (float*)d_out);
  }
}